// GATNet_39702677684841
// MI455X (gfx1250) — compile-verified
//
#include <hip/hip_runtime.h>

// ---------------------------------------------------------------------------
// GATNet forward for MI455X (gfx1250, wave32).
// GEMMs: V_WMMA_F32_16X16X4_F32, one 16x16 tile per wave, B staged in LDS
//        (shared by the 8 waves of a block), A streamed via global_load_b64.
// Edge softmax + aggregation: CDNA5 float global atomics (resolve in L2;
// h1 = 102.4MB fits in the 192MB L2, so gathers/scatters are L2-resident).
// ---------------------------------------------------------------------------

typedef __attribute__((ext_vector_type(2))) float v2f;
typedef __attribute__((ext_vector_type(8))) float v8f;

#define IN_C  128
#define HID   512
#define OUT2  64
#define NCLS  2
#define NEG_SLOPE 0.2f
#define KMAX  512            // largest K used (layer-2 GEMM)

// ---------------- WMMA f32 GEMM: C[M,N] = A[M,K] @ B[K,N] ------------------
// grid = ( ceil(tilesM/8), tilesN ), block = 256 (8 waves).
// All waves in a block share column strip tn = blockIdx.y; B[K x 16] is staged
// in LDS with interleaved pair layout: Bs[(k>>1)*32 + col*2 + (k&1)], so each
// lane's B operand (rows k+2h, k+2h+1 at column l16) is one aligned b64 read
// hitting banks {2*lane, 2*lane+1} -> conflict-free across the wave.
//
// Per ISA 7.12.2 (wave32):
//   A 16x4:  lanes 0-15 M=lane; v0 = K=0|K=2 (lane[4] selects), v1 = K=1|K=3
//   B 4x16:  v0 = row K=0 (lanes 0-15) | K=2 (lanes 16-31), v1 = K=1|K=3
//   C 16x16: VGPR r -> row (r + 8*lane[4]), col = lane&15
__global__ void gat_wmma_gemm_f32(const float* __restrict__ A,
                                  const float* __restrict__ B,
                                  float* __restrict__ C,
                                  int M, int K, int Nc) {
  __shared__ float Bs[KMAX * 16];

  const int tid  = threadIdx.x;
  const int lane = tid & 31;
  const int waveid = tid >> 5;
  const int tn = blockIdx.y;
  const int tilesM = (M + 15) >> 4;

  // cooperative, coalesced load of B[:, tn*16 .. tn*16+15] into LDS
  const int nK16 = K * 16;
  for (int idx = tid; idx < nK16; idx += 256) {
    const int k = idx >> 4;
    const int c = idx & 15;
    Bs[((k >> 1) << 5) + (c << 1) + (k & 1)] = B[(size_t)k * Nc + tn * 16 + c];
  }
  __syncthreads();

  const int tm = blockIdx.x * 8 + waveid;
  if (tm >= tilesM) return;                     // wave-uniform: EXEC stays all-1s

  const int half = lane >> 4;                   // selects K pair {0,1} vs {2,3}
  const int l16  = lane & 15;

  const float* Ap = A + (size_t)(tm * 16 + l16) * K + 2 * half;  // 8B aligned
  const float* Bl = Bs + half * 32 + (l16 << 1);                 // 8B aligned

  v8f acc = {};
#pragma unroll 8
  for (int k = 0; k < K; k += 4) {
    v2f a = *(const v2f*)(Ap + k);              // A[m][k+2h], A[m][k+2h+1]
    v2f b = *(const v2f*)(Bl + k * 16);         // B[k+2h][n], B[k+2h+1][n]
    acc = __builtin_amdgcn_wmma_f32_16x16x4_f32(
        /*neg_a=*/false, a, /*neg_b=*/false, b,
        /*c_mod=*/(short)0, acc, /*reuse_a=*/false, /*reuse_b=*/false);
  }

  float* Cp = C + (size_t)(tm * 16 + 8 * half) * Nc + tn * 16 + l16;
#pragma unroll
  for (int r = 0; r < 8; ++r) Cp[(size_t)r * Nc] = acc[r];
}

// ---------------- per-node attention dot products (wave per node) ----------
__global__ void gat_node_dots(const float* __restrict__ h,
                              const float* __restrict__ a_src,
                              const float* __restrict__ a_dst,
                              float* __restrict__ as_out,
                              float* __restrict__ ad_out,
                              int N, int D) {
  const int wave = blockIdx.x * (blockDim.x >> 5) + (threadIdx.x >> 5);
  const int lane = threadIdx.x & 31;
  if (wave >= N) return;
  const float* row = h + (size_t)wave * D;
  float ss = 0.f, sd = 0.f;
  for (int c = lane; c < D; c += 32) {
    float v = row[c];
    ss += v * a_src[c];
    sd += v * a_dst[c];
  }
#pragma unroll
  for (int off = 16; off > 0; off >>= 1) {
    ss += __shfl_down(ss, off, 32);
    sd += __shfl_down(sd, off, 32);
  }
  if (lane == 0) { as_out[wave] = ss; ad_out[wave] = sd; }
}

// ---------------- fills ----------------------------------------------------
__global__ void gat_fill_f32(float* __restrict__ p, float v, int n) {
  int i = blockIdx.x * blockDim.x + threadIdx.x;
  if (i < n) p[i] = v;
}

// ---------------- edge helpers ---------------------------------------------
__device__ __forceinline__ void edge_sd(const int* __restrict__ ei, int e,
                                        int E, int& s, int& d) {
  if (e < E) { s = ei[e]; d = ei[E + e]; }
  else       { s = d = e - E; }                 // self loops appended
}

__device__ __forceinline__ float leakyr(float v) {
  return v > 0.f ? v : NEG_SLOPE * v;
}

// float atomic max via signed/unsigned int ordering trick (init = -inf)
__device__ __forceinline__ void atomicMaxF32(float* addr, float val) {
  if (val >= 0.f) atomicMax((int*)addr, __float_as_int(val));
  else            atomicMin((unsigned int*)addr, __float_as_uint(val));
}

__global__ void gat_edge_max(const int* __restrict__ ei,
                             const float* __restrict__ as,
                             const float* __restrict__ ad,
                             float* __restrict__ m, int E, int Etot) {
  int e = blockIdx.x * blockDim.x + threadIdx.x;
  if (e >= Etot) return;
  int s, d; edge_sd(ei, e, E, s, d);
  atomicMaxF32(&m[d], leakyr(as[s] + ad[d]));
}

__global__ void gat_edge_expsum(const int* __restrict__ ei,
                                const float* __restrict__ as,
                                const float* __restrict__ ad,
                                const float* __restrict__ m,
                                float* __restrict__ alpha,
                                float* __restrict__ denom, int E, int Etot) {
  int e = blockIdx.x * blockDim.x + threadIdx.x;
  if (e >= Etot) return;
  int s, d; edge_sd(ei, e, E, s, d);
  float ex = expf(leakyr(as[s] + ad[d]) - m[d]);
  alpha[e] = ex;
  atomicAdd(&denom[d], ex);
}

__global__ void gat_edge_norm(const int* __restrict__ ei,
                              const float* __restrict__ denom,
                              float* __restrict__ alpha, int E, int Etot) {
  int e = blockIdx.x * blockDim.x + threadIdx.x;
  if (e >= Etot) return;
  int s, d; edge_sd(ei, e, E, s, d);
  alpha[e] /= denom[d];
}

// wave per edge: out[dst] += alpha[e] * h[src]   (float atomics, L2-resident)
__global__ void gat_edge_aggregate(const int* __restrict__ ei,
                                   const float* __restrict__ alpha,
                                   const float* __restrict__ h,
                                   float* __restrict__ out,
                                   int E, int Etot, int D) {
  const int wave = blockIdx.x * (blockDim.x >> 5) + (threadIdx.x >> 5);
  const int lane = threadIdx.x & 31;
  if (wave >= Etot) return;
  int s, d; edge_sd(ei, wave, E, s, d);
  const float a = alpha[wave];
  const float* hs = h + (size_t)s * D;
  float* od = out + (size_t)d * D;
  for (int c = lane; c < D; c += 32) atomicAdd(&od[c], a * hs[c]);
}

// out = relu(out + bias), in place (D must be a power of two: 512 / 64)
__global__ void gat_bias_relu(float* __restrict__ p,
                              const float* __restrict__ bias, int N, int Dmask) {
  int i = blockIdx.x * blockDim.x + threadIdx.x;
  if (i >= N) return;                           // N = total elements here
  float v = p[i] + bias[i & Dmask];
  p[i] = v > 0.f ? v : 0.f;
}

// final FC: [N,64] @ [64,2] + b
__global__ void gat_fc(const float* __restrict__ h,
                       const float* __restrict__ W,
                       const float* __restrict__ b,
                       float* __restrict__ out, int N) {
  int n = blockIdx.x * blockDim.x + threadIdx.x;
  if (n >= N) return;
  const float* row = h + (size_t)n * OUT2;
  float a0 = b[0], a1 = b[1];
#pragma unroll
  for (int c = 0; c < OUT2; ++c) {
    float v = row[c];
    a0 += v * W[c * NCLS + 0];
    a1 += v * W[c * NCLS + 1];
  }
  out[n * NCLS + 0] = a0;
  out[n * NCLS + 1] = a1;
}

// ---------------------------------------------------------------------------
extern "C" void kernel_launch(void* const* d_in, const int* in_sizes, int n_in,
                              void* d_out, int out_size, void* d_ws, size_t ws_size,
                              hipStream_t stream) {
  const float* x   = (const float*)d_in[0];
  const int*   ei  = (const int*)  d_in[1];   // [2,E] row-major: src then dst
  const float* W1  = (const float*)d_in[2];
  const float* a1s = (const float*)d_in[3];
  const float* a1d = (const float*)d_in[4];
  const float* b1  = (const float*)d_in[5];
  const float* W2  = (const float*)d_in[6];
  const float* a2s = (const float*)d_in[7];
  const float* a2d = (const float*)d_in[8];
  const float* b2  = (const float*)d_in[9];
  const float* fcW = (const float*)d_in[10];
  const float* fcb = (const float*)d_in[11];

  const int N    = in_sizes[0] / IN_C;   // 50000
  const int E    = in_sizes[1] / 2;      // 400000
  const int Etot = E + N;                // + self loops

  // workspace layout (floats)
  float* w       = (float*)d_ws;
  float* h1      = w;                                   // N*HID  (also h2 reuse)
  float* agg     = w + (size_t)N * HID;                 // N*HID  (also agg2 reuse)
  float* as_buf  = w + (size_t)2 * N * HID;             // N
  float* ad_buf  = as_buf + N;                          // N
  float* m_buf   = ad_buf + N;                          // N
  float* den_buf = m_buf + N;                           // N
  float* alpha_e = den_buf + N;                         // Etot

  const int TB = 256;
  const float NEG_INF = -__builtin_inff();

  auto gemm_grid = [](int M, int Nc) {
    return dim3((((M + 15) / 16) + 7) / 8, Nc / 16);    // 8 M-tiles per block
  };
  const int edge_blk  = (Etot + TB - 1) / TB;
  const int eagg_blk  = (Etot + 7) / 8;                 // wave per edge
  const int node_blk  = (N + 7) / 8;                    // wave per node

  // ======== layer 1 ========
  gat_wmma_gemm_f32<<<gemm_grid(N, HID), TB, 0, stream>>>(x, W1, h1, N, IN_C, HID);
  gat_node_dots<<<node_blk, TB, 0, stream>>>(h1, a1s, a1d, as_buf, ad_buf, N, HID);

  gat_fill_f32<<<(N + TB - 1) / TB, TB, 0, stream>>>(m_buf, NEG_INF, N);
  gat_fill_f32<<<(N + TB - 1) / TB, TB, 0, stream>>>(den_buf, 0.f, N);
  gat_fill_f32<<<(N * HID + TB - 1) / TB, TB, 0, stream>>>(agg, 0.f, N * HID);

  gat_edge_max   <<<edge_blk, TB, 0, stream>>>(ei, as_buf, ad_buf, m_buf, E, Etot);
  gat_edge_expsum<<<edge_blk, TB, 0, stream>>>(ei, as_buf, ad_buf, m_buf, alpha_e, den_buf, E, Etot);
  gat_edge_norm  <<<edge_blk, TB, 0, stream>>>(ei, den_buf, alpha_e, E, Etot);
  gat_edge_aggregate<<<eagg_blk, TB, 0, stream>>>(ei, alpha_e, h1, agg, E, Etot, HID);
  gat_bias_relu<<<(N * HID + TB - 1) / TB, TB, 0, stream>>>(agg, b1, N * HID, HID - 1);

  // ======== layer 2 ========  (h2 overwrites h1 buffer; agg2 reuses agg buffer)
  gat_wmma_gemm_f32<<<gemm_grid(N, OUT2), TB, 0, stream>>>(agg, W2, h1, N, HID, OUT2);
  gat_node_dots<<<node_blk, TB, 0, stream>>>(h1, a2s, a2d, as_buf, ad_buf, N, OUT2);

  gat_fill_f32<<<(N + TB - 1) / TB, TB, 0, stream>>>(m_buf, NEG_INF, N);
  gat_fill_f32<<<(N + TB - 1) / TB, TB, 0, stream>>>(den_buf, 0.f, N);
  gat_fill_f32<<<(N * OUT2 + TB - 1) / TB, TB, 0, stream>>>(agg, 0.f, N * OUT2);

  gat_edge_max   <<<edge_blk, TB, 0, stream>>>(ei, as_buf, ad_buf, m_buf, E, Etot);
  gat_edge_expsum<<<edge_blk, TB, 0, stream>>>(ei, as_buf, ad_buf, m_buf, alpha_e, den_buf, E, Etot);
  gat_edge_norm  <<<edge_blk, TB, 0, stream>>>(ei, den_buf, alpha_e, E, Etot);
  gat_edge_aggregate<<<eagg_blk, TB, 0, stream>>>(ei, alpha_e, h1, agg, E, Etot, OUT2);
  gat_bias_relu<<<(N * OUT2 + TB - 1) / TB, TB, 0, stream>>>(agg, b2, N * OUT2, OUT2 - 1);

  // ======== classifier ========
  gat_fc<<<(N + TB - 1) / TB, TB, 0, stream>>>(agg, fcW, fcb, (float*)d_out, N);
}